// ModelNew_3556232922234
// MI455X (gfx1250) — compile-verified
//
#include <hip/hip_runtime.h>

// ---------------- problem constants ----------------
#define B_      8
#define CIN     32
#define COUT    64
#define DIN     16
#define HIN     32
#define WIN     32
#define OD      31
#define OH      63
#define OW      63
#define OHW     (OH * OW)          // 3969
#define PLANE   (OD * OH * OW)     // 123039
#define NTOT    (B_ * COUT * PLANE) // 62,995,968 (divisible by 4)
#define NV4     (NTOT / 4)          // 15,748,992 float4 groups
#define XCSTR   (DIN * HIN * WIN)  // 16384 (cin stride in x)
#define EPS_    1e-5f

typedef __attribute__((ext_vector_type(2))) float v2f;
typedef __attribute__((ext_vector_type(4))) float v4f;
typedef __attribute__((ext_vector_type(8))) float v8f;

// Only check builtin availability on the DEVICE pass (host pass has no amdgcn builtins).
#if defined(__HIP_DEVICE_COMPILE__)
#if !__has_builtin(__builtin_amdgcn_wmma_f32_16x16x4_f32)
#error "gfx1250 device pass: __builtin_amdgcn_wmma_f32_16x16x4_f32 not available"
#endif
#endif

// ------------------------------------------------------------------
// Pass A: implicit-GEMM transposed conv + bias + BN, per-(b,c) sums.
// One block = (b, parity-phase, jd output d-plane). 4 waves, wave w
// owns cout [16w,16w+16). WMMA f32 16x16x4: M=cout tile, N=16 output
// positions of this phase, K = cin x taps (f32, full precision).
// ------------------------------------------------------------------
__global__ __launch_bounds__(128) void convt_bn_wmma_kernel(
    const float* __restrict__ x,     const float* __restrict__ w,
    const float* __restrict__ bias,  const float* __restrict__ rmean,
    const float* __restrict__ rvar,  const float* __restrict__ gamma,
    const float* __restrict__ beta,  float* __restrict__ out,
    float* __restrict__ sums)
{
  __shared__ float wlds[COUT * 256];   // [cout][K_TOT(+pad)] : <= 64 KB

  const int bid = blockIdx.x;
  const int jd  = bid & 15;
  const int phs = (bid >> 4) & 7;
  const int b   = bid >> 7;
  const int pd = (phs >> 2) & 1, ph = (phs >> 1) & 1, pw = phs & 1;

  const int cnt_d = pd ? 15 : 16;
  if (jd >= cnt_d) return;                       // block-uniform
  const int cnt_h = ph ? 31 : 32;
  const int cnt_w = pw ? 31 : 32;
  const int od = 2 * jd + pd;

  // tap tables per dim: parity0 -> {k=1, delta=0}; parity1 -> {k=0,d=1},{k=2,d=0}
  const int kdv[2] = {pd ? 0 : 1, 2}, ddv[2] = {pd, 0};
  const int khv[2] = {ph ? 0 : 1, 2}, dhv[2] = {ph, 0};
  const int kwv[2] = {pw ? 0 : 1, 2}, dwv[2] = {pw, 0};
  const int nh = 1 << ph, nw = 1 << pw;
  const int ntaps = 1 << (pd + ph + pw);         // 1,2,4,8
  const int ksh   = 5 + pd + ph + pw;
  const int K_TOT = 1 << ksh;                    // 32..256
  const int rowp  = K_TOT + (K_TOT < 256 ? 4 : 0); // pad rows vs LDS bank conflicts

  const int tid = threadIdx.x;

  // ---- stage weights into LDS as [cout][t*32+cin] (A-major for ds_load_b64) ----
  for (int i = tid; i < (COUT << ksh); i += 128) {
    const int cout = i >> ksh;
    const int kk   = i & (K_TOT - 1);
    const int t    = kk >> 5;
    const int cin  = kk & 31;
    const int td = t >> (ph + pw);
    const int th = (t >> pw) & (nh - 1);
    const int tw = t & (nw - 1);
    const int widx = (cin * COUT + cout) * 27 + kdv[td] * 9 + khv[th] * 3 + kwv[tw];
    wlds[cout * rowp + kk] = w[widx];
  }
  __syncthreads();

  const int lane = tid & 31;
  const int wid  = tid >> 5;
  const int co0  = wid * 16;           // wave's cout tile
  const int hi   = lane >> 4;          // K-half (A) / K-half (B)
  const int lm   = lane & 15;          // M row (A) / N col (B)
  const int cb   = co0 + hi * 8;       // first of this lane's 8 result couts

  // folded BN: y = acc*scl + off,  off = (bias - mean)*scl + beta
  float scl[8], off[8];
#pragma unroll
  for (int r = 0; r < 8; ++r) {
    const int c = cb + r;
    const float s = gamma[c] * rsqrtf(rvar[c] + EPS_);
    scl[r] = s;
    off[r] = (bias[c] - rmean[c]) * s + beta[c];
  }

  // per-tap input spatial offsets
  int doff[8];
  for (int t = 0; t < ntaps; ++t) {
    const int td = t >> (ph + pw);
    const int th = (t >> pw) & (nh - 1);
    const int tw = t & (nw - 1);
    doff[t] = ddv[td] * (HIN * WIN) + dhv[th] * WIN + dwv[tw];
  }

  const float* xb = x + b * (CIN * XCSTR);
  const int arow = (co0 + lm) * rowp;            // lane's A row base in LDS
  const int obase = (b * COUT) * PLANE + od * OHW;

  const int NP    = cnt_h * cnt_w;
  const int ntile = (NP + 15) >> 4;

  float asum[8] = {0.f, 0.f, 0.f, 0.f, 0.f, 0.f, 0.f, 0.f};

  for (int tile = 0; tile < ntile; ++tile) {
    const int nidx  = tile * 16 + lm;
    const bool valid = nidx < NP;
    const int n  = valid ? nidx : (NP - 1);      // clamp; EXEC stays all-1 for WMMA
    const int jh = n / cnt_w;
    const int jw = n - jh * cnt_w;
    const int p0 = jd * (HIN * WIN) + jh * WIN + jw;

    v8f acc = {0.f, 0.f, 0.f, 0.f, 0.f, 0.f, 0.f, 0.f};

    for (int t = 0; t < ntaps; ++t) {
      const float* xp = xb + p0 + doff[t];
      const int kkb = (t << 5) + (hi << 1);
#pragma unroll
      for (int c = 0; c < 8; ++c) {
        // A: 16x4 f32 fragment — lanes 0-15 hold K0,K1; 16-31 hold K2,K3
        const v2f a = *(const v2f*)&wlds[arow + kkb + (c << 2)];
        // B: 4x16 f32 fragment — VGPR0 row K(2*hi), VGPR1 row K(2*hi+1), col = lm
        const int cinb = (c << 2) + (hi << 1);
        v2f bm;
        bm.x = xp[cinb * XCSTR];
        bm.y = xp[(cinb + 1) * XCSTR];
        acc = __builtin_amdgcn_wmma_f32_16x16x4_f32(
            false, a, false, bm, (short)0, acc, false, false);
      }
    }

    // C layout: VGPR r -> cout = cb + r, col = lm
    const int ob = obase + (2 * jh + ph) * OW + (2 * jw + pw);
#pragma unroll
    for (int r = 0; r < 8; ++r) {
      const float v = acc[r] * scl[r] + off[r];
      if (valid) out[ob + (cb + r) * PLANE] = v;
      asum[r] += valid ? v : 0.f;
    }
  }

  // reduce across the 16 lanes sharing each cout, then one atomic per cout
#pragma unroll
  for (int r = 0; r < 8; ++r) {
    float v = asum[r];
    v += __shfl_xor(v, 8, 32);
    v += __shfl_xor(v, 4, 32);
    v += __shfl_xor(v, 2, 32);
    v += __shfl_xor(v, 1, 32);
    if (lm == 0) atomicAdd(&sums[b * COUT + cb + r], v);
  }
}

// ------------------------------------------------------------------
__global__ __launch_bounds__(256) void zero_sums_kernel(float* p, int n) {
  const int i = blockIdx.x * 256 + threadIdx.x;
  if (i < n) p[i] = 0.f;
}

// ------------------------------------------------------------------
// Pass B: y -= mean[b,c], streamed as b128 with non-temporal hints
// (504 MB of pure streaming traffic; never re-read, keep it out of L2).
// PLANE is odd, so a float4 group may straddle a (b,c) boundary: scalar
// fallback for those (384 of 15.7M groups).
// ------------------------------------------------------------------
__global__ __launch_bounds__(256) void mean_sub_vec_kernel(
    float* __restrict__ out, const float* __restrict__ sums) {
  const int i4 = blockIdx.x * 256 + threadIdx.x;
  if (i4 >= NV4) return;
  const int e0  = i4 * 4;
  const int bc0 = e0 / PLANE;
  const int bc3 = (e0 + 3) / PLANE;

  v4f v = __builtin_nontemporal_load((const v4f*)(out + e0));
  if (bc0 == bc3) {
    const float m = sums[bc0] * (1.0f / (float)PLANE);
    v.x -= m; v.y -= m; v.z -= m; v.w -= m;
  } else {
#pragma unroll
    for (int j = 0; j < 4; ++j) {
      const int bc = (e0 + j) / PLANE;
      v[j] -= sums[bc] * (1.0f / (float)PLANE);
    }
  }
  __builtin_nontemporal_store(v, (v4f*)(out + e0));
}

// ------------------------------------------------------------------
extern "C" void kernel_launch(void* const* d_in, const int* in_sizes, int n_in,
                              void* d_out, int out_size, void* d_ws, size_t ws_size,
                              hipStream_t stream) {
  const float* x     = (const float*)d_in[0];
  const float* w     = (const float*)d_in[1];
  const float* bias  = (const float*)d_in[2];
  const float* rmean = (const float*)d_in[3];
  const float* rvar  = (const float*)d_in[4];
  const float* gamma = (const float*)d_in[5];
  const float* beta  = (const float*)d_in[6];
  float* out  = (float*)d_out;
  float* sums = (float*)d_ws;          // B_*COUT floats

  zero_sums_kernel<<<(B_ * COUT + 255) / 256, 256, 0, stream>>>(sums, B_ * COUT);

  convt_bn_wmma_kernel<<<B_ * 8 * 16, 128, 0, stream>>>(
      x, w, bias, rmean, rvar, gamma, beta, out, sums);

  mean_sub_vec_kernel<<<(NV4 + 255) / 256, 256, 0, stream>>>(out, sums);
}